// StaticGraphLinear_90958817395179
// MI455X (gfx1250) — compile-verified
//
#include <hip/hip_runtime.h>

typedef float v2f __attribute__((ext_vector_type(2)));
typedef float v8f __attribute__((ext_vector_type(8)));

#define WMMA_F32X4(a, b, c) \
  __builtin_amdgcn_wmma_f32_16x16x4_f32(false, (a), false, (b), (short)0, (c), false, false)

// Problem constants (from reference)
constexpr int Bn   = 2048;
constexpr int Nn   = 64;
constexpr int DIN  = 256;
constexpr int DOUT = 256;
constexpr int Tt   = 8;

// Tiling
constexpr int BT = 16;   // b rows per workgroup
constexpr int OT = 64;   // o cols per workgroup

// LDS layout (in floats)
constexpr int GS_STRIDE = 66;                    // padded G row (bank spread)
constexpr int GS_FLOATS = 64 * GS_STRIDE;        // 4224
constexpr int PERM_OFF  = GS_FLOATS;             // 64 ints (type-sorted node order)
constexpr int XS_OFF    = PERM_OFF + 64;         // 4288
constexpr int XS_SLAB   = BT * DIN;              // 4096 floats per n-slab
constexpr int HS_OFF    = XS_OFF + 2 * XS_SLAB;  // 12480
constexpr int HB_STRIDE = (Nn / 2) * OT * 2 + 4; // 4100 (pad -> b-rows land on distinct banks)
constexpr int LDS_FLOATS = HS_OFF + BT * HB_STRIDE; // 78080 floats = 312320 B (< 320 KB)
static_assert(LDS_FLOATS * 4 <= 320 * 1024, "LDS budget");

// ---------------------------------------------------------------------------
// Prep: swizzle weight[T][DOUT][DIN] into per-lane WMMA-A layout:
//   wp[((t*16 + j)*64 + s)*32 + lane] = float2{ w[t][j*16 + (lane&15)][4s + 2*(lane>>4) + {0,1}] }
// so the main loop's A-operand loads are contiguous 256B per wave per K-step.
// ---------------------------------------------------------------------------
extern "C" __global__ __launch_bounds__(256)
void sgl_wprep(const float* __restrict__ w, float* __restrict__ wp) {
  int f = blockIdx.x * 256 + threadIdx.x;  // 0 .. 262143 (one float2 each)
  int l = f & 31;
  int s = (f >> 5) & 63;
  int j = (f >> 11) & 15;
  int t = f >> 15;
  int o  = j * 16 + (l & 15);
  int k0 = 4 * s + 2 * (l >> 4);
  v2f v = *(const v2f*)(w + ((size_t)(t * 256 + o)) * 256 + k0);
  ((v2f*)wp)[f] = v;
}

// ---------------------------------------------------------------------------
// Fused kernel: phase A computes h[b,n,o] tile into LDS with f32 WMMA,
// phase B contracts with G (and adds bias) with f32 WMMA, stores out.
// ---------------------------------------------------------------------------
extern "C" __global__ __launch_bounds__(256)
void sgl_fused(const float* __restrict__ x, const float* __restrict__ G,
               const float* __restrict__ wprep, const float* __restrict__ bias,
               const int* __restrict__ ntypes, float* __restrict__ out) {
  extern __shared__ float lds[];
  const int tid  = threadIdx.x;
  const int lane = tid & 31;
  const int wave = tid >> 5;
  const int b0 = blockIdx.x * BT;
  const int otile = blockIdx.y;       // 0..3
  const int o0 = otile * OT;

  // ---- stage G (64x64) into LDS with padded rows ----
  for (int i = tid; i < 64 * 64; i += 256) {
    int m = i >> 6, n = i & 63;
    lds[m * GS_STRIDE + n] = G[i];
  }
  // ---- type-sorted permutation of node ids (L0 reuse of weight streams) ----
  if (tid < 64) {
    int t = ntypes[tid];
    int rank = 0;
    for (int m2 = 0; m2 < 64; ++m2) {
      int tm = ntypes[m2];
      rank += (tm < t) || (tm == t && m2 < tid);
    }
    ((int*)lds)[PERM_OFF + rank] = tid;
  }
  __syncthreads();

  const int g     = wave >> 2;         // which n-slab this wave consumes (0/1)
  const int osub  = (wave & 3) * 16;   // o sub-tile within OT
  const int jglob = otile * 4 + (wave & 3);

  // =========================== Phase A ===========================
  // h[b, n, o0..o0+63] for all 64 nodes, 2 nodes per iteration.
  for (int it = 0; it < Nn / 2; ++it) {
    // Cooperative load of 2 x-slabs, transposed into K-pair layout:
    // element (b,k) -> dword (k>>1)*32 + b*2 + (k&1)
    {
      int slab = tid >> 7;    // 0/1
      int u  = tid & 127;
      int b  = u & 15;
      int kc = u >> 4;        // 0..7 -> covers 32 k each
      int n = ((int*)lds)[PERM_OFF + it * 2 + slab];
      const float* xs = x + ((size_t)(b0 + b) * Nn + n) * DIN + kc * 32;
      float* dst = lds + XS_OFF + slab * XS_SLAB;
#pragma unroll
      for (int q = 0; q < 8; ++q) {
        float4 v = ((const float4*)xs)[q];
        int k = kc * 32 + q * 4;
        int p = k >> 1;
        ((v2f*)dst)[p * 16 + b]       = v2f{v.x, v.y};
        ((v2f*)dst)[(p + 1) * 16 + b] = v2f{v.z, v.w};
      }
    }
    __syncthreads();

    // One 16(o) x 16(b) tile per wave, K = DIN = 256 -> 64 WMMAs.
    // Two independent accumulators (even/odd K-step) break the D->C RAW chain
    // so the XDL pipe can overlap back-to-back WMMAs.
    {
      int n = ((int*)lds)[PERM_OFF + it * 2 + g];
      int t = ntypes[n];  // wave-uniform -> scalar load
      const v2f* wa = (const v2f*)wprep + (size_t)((t * 16 + jglob) * 64) * 32 + lane;
      const float* xb = lds + XS_OFF + g * XS_SLAB;
      v8f c0 = {}, c1 = {};
#pragma unroll 4
      for (int s = 0; s < 64; s += 2) {
        v2f a0 = wa[s * 32];                                                   // A: w (M=o,K)
        v2f a1 = wa[(s + 1) * 32];
        v2f e0 = *(const v2f*)(xb + ((2 * s + (lane >> 4)) * 32 + (lane & 15) * 2));       // B: x^T
        v2f e1 = *(const v2f*)(xb + ((2 * (s + 1) + (lane >> 4)) * 32 + (lane & 15) * 2));
        c0 = WMMA_F32X4(a0, e0, c0);
        c1 = WMMA_F32X4(a1, e1, c1);
      }
      v8f c = c0 + c1;
      // D tile: lane holds N=b=(lane&15); VGPR r holds M=o = osub + r + 8*(lane>>4)
      int b = lane & 15;
      int base = HS_OFF + b * HB_STRIDE + ((n >> 1) * OT) * 2 + (n & 1);
      int ob = osub + 8 * (lane >> 4);
#pragma unroll
      for (int r = 0; r < 8; ++r)
        lds[base + (ob + r) * 2] = c[r];
    }
    __syncthreads();
  }

  // =========================== Phase B ===========================
  // out[b, m, o] = sum_n G[m,n] * h[b,n,o] + bias[o]
  // jobs: 16 b x 4 m-subtiles x 4 o-subtiles = 256, K = 64 -> 16 WMMAs each,
  // split across two accumulators.
  for (int job = wave; job < 256; job += 8) {
    int os = (job & 3) * 16;
    int m0 = ((job >> 2) & 3) * 16;
    int b  = job >> 4;
    const float* ga = lds + (m0 + (lane & 15)) * GS_STRIDE + 2 * (lane >> 4);
    const float* hb = lds + HS_OFF + b * HB_STRIDE + (lane >> 4) * OT * 2 +
                      (os + (lane & 15)) * 2;
    v8f c0 = {}, c1 = {};
#pragma unroll
    for (int s = 0; s < 16; s += 2) {
      v2f a0 = *(const v2f*)(ga + 4 * s);              // A: G (M=m, K=n)
      v2f a1 = *(const v2f*)(ga + 4 * (s + 1));
      v2f e0 = *(const v2f*)(hb + s * 4 * OT);         // B: h (K=n, N=o), pair-contiguous
      v2f e1 = *(const v2f*)(hb + (s + 1) * 4 * OT);
      c0 = WMMA_F32X4(a0, e0, c0);
      c1 = WMMA_F32X4(a1, e1, c1);
    }
    v8f c = c0 + c1;
    int oglob = o0 + os + (lane & 15);
    float bv = bias[oglob];
    float* op = out + ((size_t)(b0 + b) * Nn) * DOUT + oglob;
#pragma unroll
    for (int r = 0; r < 8; ++r) {
      int m = m0 + r + 8 * (lane >> 4);
      op[(size_t)m * DOUT] = c[r] + bv;
    }
  }
}

// ---------------------------------------------------------------------------
extern "C" void kernel_launch(void* const* d_in, const int* in_sizes, int n_in,
                              void* d_out, int out_size, void* d_ws, size_t ws_size,
                              hipStream_t stream) {
  const float* x    = (const float*)d_in[0];
  const float* G    = (const float*)d_in[1];
  const float* w    = (const float*)d_in[2];
  const float* bias = (const float*)d_in[3];
  const int*   nt   = (const int*)d_in[4];
  float* out = (float*)d_out;
  float* wp  = (float*)d_ws;  // 2 MB swizzled weights

  // Weight swizzle: 262144 float2's
  sgl_wprep<<<1024, 256, 0, stream>>>(w, wp);

  size_t shbytes = (size_t)LDS_FLOATS * sizeof(float);
  (void)hipFuncSetAttribute((const void*)sgl_fused,
                            hipFuncAttributeMaxDynamicSharedMemorySize,
                            (int)shbytes);
  dim3 grid(Bn / BT, DOUT / OT);
  sgl_fused<<<grid, 256, shbytes, stream>>>(x, G, wp, bias, nt, out);
}